// PcheckDynamicBalancedMarginInnerProduct_67765993997130
// MI455X (gfx1250) — compile-verified
//
#include <hip/hip_runtime.h>
#include <hip/hip_bf16.h>
#include <math.h>

#define FEATURE_DIM 512
#define CLASS_NUM   10572
#define BATCH       2048
#define SCALEF      30.0f
#define PI_F        3.14159265358979323846f

typedef __attribute__((ext_vector_type(2))) float v2f;
typedef __attribute__((ext_vector_type(8))) float v8f;

// ---------------------------------------------------------------------------
// CDNA5 async global->LDS copy (ASYNCcnt-tracked DMA into LDS, no VGPR data).
// lds_off = byte offset into workgroup LDS (low 32 bits of flat shared ptr).
// ---------------------------------------------------------------------------
__device__ __forceinline__ void async_copy16(unsigned lds_off, const float* gsrc) {
    asm volatile("global_load_async_to_lds_b128 %0, %1, off"
                 :: "v"(lds_off), "v"(gsrc)
                 : "memory");
}

// ---------------------------------------------------------------------------
// Row L2-normalize: one block (256 threads) per row of length 512.
// ---------------------------------------------------------------------------
__global__ __launch_bounds__(256) void rownorm_kernel(const float* __restrict__ in,
                                                      float* __restrict__ out) {
    const int row = blockIdx.x;
    const int t   = threadIdx.x;
    const float* src = in  + (size_t)row * FEATURE_DIM;
    float*       dst = out + (size_t)row * FEATURE_DIM;
    float x0 = src[t];
    float x1 = src[t + 256];
    __shared__ float red[256];
    red[t] = x0 * x0 + x1 * x1;
    __syncthreads();
    for (int s = 128; s > 0; s >>= 1) {
        if (t < s) red[t] += red[t + s];
        __syncthreads();
    }
    const float inv = rsqrtf(red[0]);
    dst[t]       = x0 * inv;
    dst[t + 256] = x1 * inv;
}

// ---------------------------------------------------------------------------
// WMMA fp32 GEMM: C[B x N] = A[B x K] * W[N x K]^T  (cos matrix)
// Block: 256 threads = 8 waves as 2(M) x 4(N). Block tile 32 x 256.
// Each wave: 16 rows x 64 cols = 4 accumulators of V_WMMA_F32_16X16X4_F32.
// Double-buffered LDS staged by GLOBAL_LOAD_ASYNC_TO_LDS_B128 (ASYNCcnt).
// LDS row stride = 36 floats (144 B): 16B-aligned for B128 async writes and
// conflict-free (36*c mod 64 distinct for c=0..15).
// ---------------------------------------------------------------------------
#define KC   32
#define LDP  36
#define MBLK 32
#define NBLK 256

__global__ __launch_bounds__(256) void gemm_wmma_kernel(const float* __restrict__ A,
                                                        const float* __restrict__ W,
                                                        float* __restrict__ C) {
    __shared__ __align__(16) float sA[2][MBLK * LDP];
    __shared__ __align__(16) float sB[2][NBLK * LDP];

    const int tid  = threadIdx.x;
    const int wave = tid >> 5;
    const int lane = tid & 31;
    const int half = lane >> 4;   // 0|1
    const int r    = lane & 15;   // 0..15
    const int wm   = wave >> 2;   // 0..1
    const int wn   = wave & 3;    // 0..3

    const int mbase = blockIdx.y * MBLK;
    const int nbase = blockIdx.x * NBLK;
    const int m0    = wm * 16;
    const int n0    = wn * 64;

    v8f acc[4] = {};

    // Cooperative staging indices
    const int a_row = tid >> 3;          // 0..31
    const int a_k4  = (tid & 7) << 2;    // 0..28 step 4
    const int b_col = tid;               // 0..255
    const int b_c   = nbase + b_col;
    const bool b_ok = (b_c < CLASS_NUM);
    const int b_cs  = b_ok ? b_c : 0;    // clamped: keep async-issue count uniform

    const float* aSrc = A + (size_t)(mbase + a_row) * FEATURE_DIM + a_k4;
    const float* bSrc = W + (size_t)b_cs * FEATURE_DIM;

    // Issue one stage's 9 async B128 copies (1 for A, 8 for B) per thread.
    auto stage = [&](int buf, int k0) {
        async_copy16((unsigned)(uintptr_t)&sA[buf][a_row * LDP + a_k4], aSrc + k0);
        const float* wp = bSrc + k0;
#pragma unroll
        for (int i = 0; i < 8; ++i) {
            async_copy16((unsigned)(uintptr_t)&sB[buf][b_col * LDP + 4 * i], wp + 4 * i);
        }
    };

    stage(0, 0);

    for (int k0 = 0; k0 < FEATURE_DIM; k0 += KC) {
        const int  buf  = (k0 / KC) & 1;
        const bool more = (k0 + KC) < FEATURE_DIM;

        if (more) {
            stage(buf ^ 1, k0 + KC);
            // 9 asyncs of the next stage may stay in flight; in-order completion
            // means <=9 outstanding guarantees the current stage has landed.
            asm volatile("s_wait_asynccnt 0x9" ::: "memory");
        } else {
            asm volatile("s_wait_asynccnt 0x0" ::: "memory");
        }

        // Zero out-of-range B columns (written after their asyncs completed).
        if (!b_ok) {
            float* bp = &sB[buf][b_col * LDP];
#pragma unroll
            for (int i = 0; i < KC; ++i) bp[i] = 0.0f;
        }
        __syncthreads();

        // ---- compute: 8 k-steps x 4 N-tiles of V_WMMA_F32_16X16X4_F32 ----
        const float* aBase = &sA[buf][(m0 + r) * LDP];
#pragma unroll
        for (int kk = 0; kk < KC; kk += 4) {
            const int ko = kk + 2 * half;
            v2f a;
            a.x = aBase[ko];
            a.y = aBase[ko + 1];
#pragma unroll
            for (int nt = 0; nt < 4; ++nt) {
                const float* bBase = &sB[buf][(n0 + nt * 16 + r) * LDP];
                v2f b;
                b.x = bBase[ko];
                b.y = bBase[ko + 1];
                acc[nt] = __builtin_amdgcn_wmma_f32_16x16x4_f32(
                    false, a, false, b, (short)0, acc[nt], false, false);
            }
        }
        __syncthreads();
    }

    // ---- store: acc VGPR j -> row (m0 + 8*half + j), lane r -> col ----
    const int row0 = mbase + m0 + half * 8;
#pragma unroll
    for (int nt = 0; nt < 4; ++nt) {
        const int col = nbase + n0 + nt * 16 + r;
        if (col < CLASS_NUM) {
#pragma unroll
            for (int j = 0; j < 8; ++j) {
                C[(size_t)(row0 + j) * CLASS_NUM + col] = acc[nt][j];
            }
        }
    }
}

// ---------------------------------------------------------------------------
// Margin stats: gather cos_gt, reduce theta max/min/avg, per-row margins.
// Single block, 1024 threads, 2 rows each.
// ---------------------------------------------------------------------------
__global__ __launch_bounds__(1024) void margin_kernel(const float* __restrict__ cosm,
                                                      const long long* __restrict__ label,
                                                      float* __restrict__ cos_gt,
                                                      float* __restrict__ add_margin,
                                                      float* __restrict__ ext_val) {
    const int t = threadIdx.x;
    __shared__ float smax[1024], smin[1024], ssum[1024];

    float th[2], cg[2];
    float mx = -1e30f, mn = 1e30f, sm = 0.0f;
#pragma unroll
    for (int i = 0; i < 2; ++i) {
        const int b = t + i * 1024;
        const int lb = (int)label[b];
        const float c = cosm[(size_t)b * CLASS_NUM + lb];
        cg[i] = c;
        const float theta = acosf(c) * (180.0f / PI_F);
        th[i] = theta;
        mx = fmaxf(mx, theta);
        mn = fminf(mn, theta);
        sm += theta;
    }
    smax[t] = mx; smin[t] = mn; ssum[t] = sm;
    __syncthreads();
    for (int s = 512; s > 0; s >>= 1) {
        if (t < s) {
            smax[t] = fmaxf(smax[t], smax[t + s]);
            smin[t] = fminf(smin[t], smin[t + s]);
            ssum[t] += ssum[t + s];
        }
        __syncthreads();
    }
    const float max_t = smax[0];
    const float min_t = smin[0];
    const float avg_t = ssum[0] * (1.0f / BATCH);
    const float m_hi  = (max_t < 90.0f) ? (90.0f - avg_t) * (PI_F / 180.0f) : 0.0f;
    const float m_lo  = min_t * (PI_F / 180.0f);

#pragma unroll
    for (int i = 0; i < 2; ++i) {
        const int b = t + i * 1024;
        const float marg   = (th[i] > avg_t) ? m_hi : m_lo;
        const bool  inside = cg[i] > -cosf(marg);
        cos_gt[b]     = cg[i];
        add_margin[b] = inside ? marg : 0.0f;
        ext_val[b]    = inside ? 0.0f : (-marg * sinf(marg));
    }
}

// ---------------------------------------------------------------------------
// marginal_logits bulk path: mlog = SCALE * cos (vectorized), then fixup.
// ---------------------------------------------------------------------------
__global__ __launch_bounds__(256) void scale_copy_kernel(const float4* __restrict__ in,
                                                         float4* __restrict__ out, int n4) {
    const int i = blockIdx.x * 256 + threadIdx.x;
    if (i < n4) {
        float4 v = in[i];
        v.x *= SCALEF; v.y *= SCALEF; v.z *= SCALEF; v.w *= SCALEF;
        out[i] = v;
    }
}

__global__ __launch_bounds__(256) void fixup_kernel(float* __restrict__ mlog,
                                                    const long long* __restrict__ label,
                                                    const float* __restrict__ cos_gt,
                                                    const float* __restrict__ add_margin,
                                                    const float* __restrict__ ext_val) {
    const int b = blockIdx.x * 256 + threadIdx.x;
    if (b < BATCH) {
        const int c = (int)label[b];
        mlog[(size_t)b * CLASS_NUM + c] =
            SCALEF * (cosf(acosf(cos_gt[b]) + add_margin[b]) + ext_val[b]);
    }
}

// ---------------------------------------------------------------------------
// Per-row softmax stats: row max, sum-exp, value at column label[-1].
// gt_probs[b] = e^{l-m}/S ;  Bs[b] = e^{m} * (S - e^{l-m}).
// ---------------------------------------------------------------------------
__global__ __launch_bounds__(256) void rowstats_kernel(const float* __restrict__ cosm,
                                                       const long long* __restrict__ label,
                                                       float* __restrict__ gt_probs,
                                                       float* __restrict__ Bs) {
    const int b = blockIdx.x;
    const int t = threadIdx.x;
    const int ll = (int)label[BATCH - 1];
    const float* row = cosm + (size_t)b * CLASS_NUM;

    __shared__ float red[256];
    float mx = -1e30f;
    for (int c = t; c < CLASS_NUM; c += 256) mx = fmaxf(mx, row[c]);
    red[t] = mx;
    __syncthreads();
    for (int s = 128; s > 0; s >>= 1) {
        if (t < s) red[t] = fmaxf(red[t], red[t + s]);
        __syncthreads();
    }
    const float m = SCALEF * red[0];
    __syncthreads();

    float se = 0.0f;
    for (int c = t; c < CLASS_NUM; c += 256) se += expf(SCALEF * row[c] - m);
    red[t] = se;
    __syncthreads();
    for (int s = 128; s > 0; s >>= 1) {
        if (t < s) red[t] += red[t + s];
        __syncthreads();
    }
    if (t == 0) {
        const float S  = red[0];
        const float e  = expf(SCALEF * row[ll] - m);
        gt_probs[b] = e / S;
        Bs[b]       = expf(m) * (S - e);
    }
}

// ---------------------------------------------------------------------------
// Final scalars: avg/min/max of gt_probs, mean of Bs -> B_avg.
// ---------------------------------------------------------------------------
__global__ __launch_bounds__(1024) void final_kernel(const float* __restrict__ gt_probs,
                                                     const float* __restrict__ Bs,
                                                     float* __restrict__ out4) {
    const int t = threadIdx.x;
    __shared__ float smax[1024], smin[1024], ssum[1024], sbs[1024];
    const float a0 = gt_probs[t], a1 = gt_probs[t + 1024];
    smax[t] = fmaxf(a0, a1);
    smin[t] = fminf(a0, a1);
    ssum[t] = a0 + a1;
    sbs[t]  = Bs[t] + Bs[t + 1024];
    __syncthreads();
    for (int s = 512; s > 0; s >>= 1) {
        if (t < s) {
            smax[t] = fmaxf(smax[t], smax[t + s]);
            smin[t] = fminf(smin[t], smin[t + s]);
            ssum[t] += ssum[t + s];
            sbs[t]  += sbs[t + s];
        }
        __syncthreads();
    }
    if (t == 0) {
        const float avg_p = ssum[0] * (1.0f / BATCH);
        const float b_mean = sbs[0] * (1.0f / BATCH);
        const float b_avg = acosf(logf(b_mean / (float)(CLASS_NUM - 1)) / SCALEF)
                            * (180.0f / PI_F);
        out4[0] = avg_p;
        out4[1] = smin[0];
        out4[2] = smax[0];
        out4[3] = b_avg;
    }
}

// ---------------------------------------------------------------------------
// Launch
// ---------------------------------------------------------------------------
extern "C" void kernel_launch(void* const* d_in, const int* in_sizes, int n_in,
                              void* d_out, int out_size, void* d_ws, size_t ws_size,
                              hipStream_t stream) {
    (void)in_sizes; (void)n_in; (void)out_size; (void)ws_size;

    const float*     feat    = (const float*)d_in[0];
    const long long* label   = (const long long*)d_in[1];
    const float*     weights = (const float*)d_in[2];

    float* out  = (float*)d_out;
    float* cosm = out;                                     // BATCH*CLASS_NUM
    float* mlog = out + (size_t)BATCH * CLASS_NUM;         // BATCH*CLASS_NUM
    float* sc4  = mlog + (size_t)BATCH * CLASS_NUM;        // 4 scalars

    float* ws        = (float*)d_ws;
    float* nf        = ws;                                       // 2048*512
    float* nw        = nf + (size_t)BATCH * FEATURE_DIM;         // 10572*512
    float* cos_gt    = nw + (size_t)CLASS_NUM * FEATURE_DIM;     // 2048
    float* add_marg  = cos_gt + BATCH;                           // 2048
    float* ext_val   = add_marg + BATCH;                         // 2048
    float* gt_probs  = ext_val + BATCH;                          // 2048
    float* Bsv       = gt_probs + BATCH;                         // 2048

    // 1) normalize rows
    rownorm_kernel<<<BATCH, 256, 0, stream>>>(feat, nf);
    rownorm_kernel<<<CLASS_NUM, 256, 0, stream>>>(weights, nw);

    // 2) WMMA fp32 GEMM -> cos
    dim3 grid((CLASS_NUM + NBLK - 1) / NBLK, BATCH / MBLK);
    gemm_wmma_kernel<<<grid, 256, 0, stream>>>(nf, nw, cosm);

    // 3) margin statistics
    margin_kernel<<<1, 1024, 0, stream>>>(cosm, label, cos_gt, add_marg, ext_val);

    // 4) marginal logits = 30*cos, then per-row fixup at label column
    const int n4 = (BATCH * CLASS_NUM) / 4;
    scale_copy_kernel<<<(n4 + 255) / 256, 256, 0, stream>>>((const float4*)cosm,
                                                            (float4*)mlog, n4);
    fixup_kernel<<<(BATCH + 255) / 256, 256, 0, stream>>>(mlog, label, cos_gt,
                                                          add_marg, ext_val);

    // 5) per-row softmax stats
    rowstats_kernel<<<BATCH, 256, 0, stream>>>(cosm, label, gt_probs, Bsv);

    // 6) final scalars
    final_kernel<<<1, 1024, 0, stream>>>(gt_probs, Bsv, sc4);
}